// GINModel_33560874451042
// MI455X (gfx1250) — compile-verified
//
#include <hip/hip_runtime.h>

typedef __attribute__((ext_vector_type(2))) float v2f;
typedef __attribute__((ext_vector_type(8))) float v8f;

#define N_NODES 100000
#define N_EDGES 1600000

// out[i] = s * in[i], vectorized float4
__global__ void scale_copy_kernel(const float* __restrict__ in, float* __restrict__ out,
                                  int n4, float s) {
    int i = blockIdx.x * blockDim.x + threadIdx.x;
    if (i >= n4) return;
    float4 v = ((const float4*)in)[i];
    float4 o;
    o.x = v.x * s; o.y = v.y * s; o.z = v.z * s; o.w = v.w * s;
    ((float4*)out)[i] = o;
}

// out[dst[e]] += feat[src[e]]  (F floats per row), float4 per thread,
// atomic f32 adds resolved in L2 (entire feature matrix fits in 192MB L2).
template<int F>
__global__ void scatter_add_kernel(const float* __restrict__ feat,
                                   const int* __restrict__ sidx,
                                   const int* __restrict__ didx,
                                   float* __restrict__ out, int nE) {
    constexpr int CH = F / 4;           // float4 chunks per edge
    int tid = blockIdx.x * blockDim.x + threadIdx.x;
    if (tid >= nE * CH) return;
    int e = tid / CH;
    int c = tid - e * CH;
    int s = sidx[e];
    int d = didx[e];
    const float4 v = *(const float4*)(feat + (size_t)s * F + c * 4);
    float* o = out + (size_t)d * F + c * 4;
    unsafeAtomicAdd(o + 0, v.x);
    unsafeAtomicAdd(o + 1, v.y);
    unsafeAtomicAdd(o + 2, v.z);
    unsafeAtomicAdd(o + 3, v.w);
}

// out[N x M] = act(A[N x K] @ W[M x K]^T + bias), fp32 WMMA 16x16x4.
// One wave computes a 16x32 output tile (two 16x16 accumulators, A reused).
// blockDim.x = 32 * (M/32) waves cover all M columns; gridDim.x = N/16 row tiles.
template<int K, int M, bool RELU>
__global__ void gemm_wmma_kernel(const float* __restrict__ A,
                                 const float* __restrict__ W,
                                 const float* __restrict__ bias,
                                 float* __restrict__ out) {
    const int lane = threadIdx.x & 31;
    const int wave = threadIdx.x >> 5;
    const int li   = lane & 15;   // row/col index within fragment
    const int kh   = lane >> 4;   // K-half select (0 or 1)
    const int mtile = blockIdx.x * 16;
    const int n0    = wave * 32;

    // A fragment: lane l holds A[mtile + l%16][kb + 2*(l>>4) + {0,1}]
    const float* arow = A + (size_t)(mtile + li) * K + 2 * kh;
    // B = W^T: lane l holds W[n + l%16][kb + 2*(l>>4) + {0,1}]  (contiguous in K)
    const float* w0 = W + (size_t)(n0 + li) * K + 2 * kh;
    const float* w1 = W + (size_t)(n0 + 16 + li) * K + 2 * kh;

    v8f c0 = {};
    v8f c1 = {};
#pragma unroll
    for (int kb = 0; kb < K; kb += 4) {
        v2f a  = *(const v2f*)(arow + kb);
        v2f b0 = *(const v2f*)(w0 + kb);
        v2f b1 = *(const v2f*)(w1 + kb);
        // (neg_a, A, neg_b, B, c_mod, C, reuse_a, reuse_b)
        c0 = __builtin_amdgcn_wmma_f32_16x16x4_f32(false, a, false, b0,
                                                   (short)0, c0, false, false);
        c1 = __builtin_amdgcn_wmma_f32_16x16x4_f32(false, a, false, b1,
                                                   (short)0, c1, false, false);
    }

    const float bs0 = bias[n0 + li];
    const float bs1 = bias[n0 + 16 + li];
    // D layout: VGPR r, lanes 0-15 -> row r, lanes 16-31 -> row r+8; col = lane%16
    const int mbase = mtile + kh * 8;
#pragma unroll
    for (int r = 0; r < 8; ++r) {
        float v0 = c0[r] + bs0;
        float v1 = c1[r] + bs1;
        if (RELU) { v0 = fmaxf(v0, 0.0f); v1 = fmaxf(v1, 0.0f); }
        out[(size_t)(mbase + r) * M + n0 + li]      = v0;
        out[(size_t)(mbase + r) * M + n0 + 16 + li] = v1;
    }
}

extern "C" void kernel_launch(void* const* d_in, const int* in_sizes, int n_in,
                              void* d_out, int out_size, void* d_ws, size_t ws_size,
                              hipStream_t stream) {
    const float* x   = (const float*)d_in[0];
    const int*   ei  = (const int*)d_in[1];
    const int*   src = ei;            // edge_index[0]
    const int*   dst = ei + N_EDGES;  // edge_index[1]
    const float* w11 = (const float*)d_in[2];
    const float* b11 = (const float*)d_in[3];
    const float* w12 = (const float*)d_in[4];
    const float* b12 = (const float*)d_in[5];
    const float* w21 = (const float*)d_in[6];
    const float* b21 = (const float*)d_in[7];
    const float* w22 = (const float*)d_in[8];
    const float* b22 = (const float*)d_in[9];
    float* out = (float*)d_out;

    // Workspace layout (floats): agg1[N*128] | bufA[N*256] | bufB[N*256] = 256MB
    float* ws   = (float*)d_ws;
    float* agg1 = ws;
    float* bufA = agg1 + (size_t)N_NODES * 128;  // h1, then agg2
    float* bufB = bufA + (size_t)N_NODES * 256;  // g1, then h2

    const int ROW_TILES = N_NODES / 16;  // 6250

    // ---- Layer 1 ----
    // agg1 = (1+eps) * x
    {
        int n4 = N_NODES * 128 / 4;
        scale_copy_kernel<<<(n4 + 255) / 256, 256, 0, stream>>>(x, agg1, n4, 1.1f);
    }
    // agg1 += segment_sum(x[src] -> dst)
    {
        int total = N_EDGES * (128 / 4);
        scatter_add_kernel<128><<<(total + 255) / 256, 256, 0, stream>>>(
            x, src, dst, agg1, N_EDGES);
    }
    // h1 = relu(agg1 @ w11^T + b11)   [N x 256]
    gemm_wmma_kernel<128, 256, true><<<ROW_TILES, 256, 0, stream>>>(agg1, w11, b11, bufA);
    // g1 = relu(h1 @ w12^T + b12)     [N x 256]  (outer ReLU fused)
    gemm_wmma_kernel<256, 256, true><<<ROW_TILES, 256, 0, stream>>>(bufA, w12, b12, bufB);

    // ---- Layer 2 ----
    // agg2 = (1+eps) * g1   (reuses bufA; h1 is dead)
    {
        int n4 = N_NODES * 256 / 4;
        scale_copy_kernel<<<(n4 + 255) / 256, 256, 0, stream>>>(bufB, bufA, n4, 1.1f);
    }
    // agg2 += segment_sum(g1[src] -> dst)
    {
        int total = N_EDGES * (256 / 4);
        scatter_add_kernel<256><<<(total + 255) / 256, 256, 0, stream>>>(
            bufB, src, dst, bufA, N_EDGES);
    }
    // h2 = relu(agg2 @ w21^T + b21)   [N x 256]  (reuses bufB; g1 is dead)
    gemm_wmma_kernel<256, 256, true><<<ROW_TILES, 256, 0, stream>>>(bufA, w21, b21, bufB);
    // out = h2 @ w22^T + b22          [N x 64]
    gemm_wmma_kernel<256, 64, false><<<ROW_TILES, 64, 0, stream>>>(bufB, w22, b22, out);
}